// ZPressor_42554535969621
// MI455X (gfx1250) — compile-verified
//
#include <hip/hip_runtime.h>
#include <hip/hip_bf16.h>

typedef __attribute__((ext_vector_type(2))) float v2f;
typedef __attribute__((ext_vector_type(8))) float v8f;

#define FPS_THREADS 512
#define NPOINT 64

// ---------------------------------------------------------------------------
// Kernel 1: farthest point sampling. One block per batch. Distance array in
// global scratch (L2-resident at 400KB/batch). Also emits centroid_xyz,
// centroid indices (as float), and the WMMA A-tile data (-2cx,-2cy,-2cz,|c|^2).
// ---------------------------------------------------------------------------
__global__ __launch_bounds__(FPS_THREADS)
void fps_kernel(const float* __restrict__ xyz, float* __restrict__ out,
                float* __restrict__ ws, int N, int B) {
  const int b  = blockIdx.x;
  const int t  = threadIdx.x;
  const int nt = blockDim.x;

  const float* __restrict__ p    = xyz + (size_t)b * N * 3;
  float* __restrict__ dist       = ws + (size_t)b * N;
  float* __restrict__ cdata      = ws + (size_t)B * N + (size_t)b * NPOINT * 4;
  float* __restrict__ out_cxyz   = out + (size_t)b * NPOINT * 3;
  float* __restrict__ out_cidx   = out + (size_t)B * NPOINT * 3 + (size_t)b * NPOINT;

  __shared__ float s_val[FPS_THREADS];
  __shared__ int   s_idx[FPS_THREADS];
  __shared__ float s_cent[3];
  __shared__ int   s_far;

  // ---- barycenter (mean over N) ----
  float sx = 0.f, sy = 0.f, sz = 0.f;
  for (int i = t; i < N; i += nt) {
    size_t o = 3 * (size_t)i;
    sx += p[o + 0]; sy += p[o + 1]; sz += p[o + 2];
  }
  s_val[t] = sx; __syncthreads();
  for (int s = nt >> 1; s > 0; s >>= 1) { if (t < s) s_val[t] += s_val[t + s]; __syncthreads(); }
  if (t == 0) s_cent[0] = s_val[0] / (float)N;
  __syncthreads();
  s_val[t] = sy; __syncthreads();
  for (int s = nt >> 1; s > 0; s >>= 1) { if (t < s) s_val[t] += s_val[t + s]; __syncthreads(); }
  if (t == 0) s_cent[1] = s_val[0] / (float)N;
  __syncthreads();
  s_val[t] = sz; __syncthreads();
  for (int s = nt >> 1; s > 0; s >>= 1) { if (t < s) s_val[t] += s_val[t + s]; __syncthreads(); }
  if (t == 0) s_cent[2] = s_val[0] / (float)N;
  __syncthreads();

  // ---- initial farthest = argmax dist-to-barycenter; init dist = 1e10 ----
  {
    const float bx = s_cent[0], by = s_cent[1], bz = s_cent[2];
    float bv = -1.f; int bi = 0;
    for (int i = t; i < N; i += nt) {
      size_t o = 3 * (size_t)i;
      float dx = p[o] - bx, dy = p[o + 1] - by, dz = p[o + 2] - bz;
      float d = dx * dx + dy * dy + dz * dz;
      dist[i] = 1e10f;
      // per-thread indices strictly increase -> strict '>' preserves first-max
      bi = (d > bv) ? i : bi;
      bv = fmaxf(bv, d);
    }
    s_val[t] = bv; s_idx[t] = bi; __syncthreads();
    for (int s = nt >> 1; s > 0; s >>= 1) {
      if (t < s) {
        float v0 = s_val[t], v2 = s_val[t + s];
        int   i0 = s_idx[t], i2 = s_idx[t + s];
        // non-short-circuit -> selects, no EXEC-mask branching
        bool take = (v2 > v0) | ((v2 == v0) & (i2 < i0));
        s_val[t] = take ? v2 : v0;
        s_idx[t] = take ? i2 : i0;
      }
      __syncthreads();
    }
    if (t == 0) s_far = s_idx[0];
    __syncthreads();
  }

  // ---- main FPS loop ----
  for (int k = 0; k < NPOINT; ++k) {
    if (t == 0) {
      int f = s_far;
      size_t o = 3 * (size_t)f;
      float cx = p[o], cy = p[o + 1], cz = p[o + 2];
      s_cent[0] = cx; s_cent[1] = cy; s_cent[2] = cz;
      out_cidx[k] = (float)f;
      out_cxyz[3 * k + 0] = cx; out_cxyz[3 * k + 1] = cy; out_cxyz[3 * k + 2] = cz;
      cdata[4 * k + 0] = -2.f * cx;
      cdata[4 * k + 1] = -2.f * cy;
      cdata[4 * k + 2] = -2.f * cz;
      cdata[4 * k + 3] = cx * cx + cy * cy + cz * cz;
    }
    __syncthreads();
    const float cx = s_cent[0], cy = s_cent[1], cz = s_cent[2];
    float bv = -1.f; int bi = 0;
    for (int i = t; i < N; i += nt) {
      size_t o = 3 * (size_t)i;
      float dx = p[o] - cx, dy = p[o + 1] - cy, dz = p[o + 2] - cz;
      float d  = dx * dx + dy * dy + dz * dz;
      float nd = fminf(dist[i], d);
      dist[i] = nd;
      bi = (nd > bv) ? i : bi;
      bv = fmaxf(bv, nd);
    }
    s_val[t] = bv; s_idx[t] = bi; __syncthreads();
    for (int s = nt >> 1; s > 0; s >>= 1) {
      if (t < s) {
        float v0 = s_val[t], v2 = s_val[t + s];
        int   i0 = s_idx[t], i2 = s_idx[t + s];
        bool take = (v2 > v0) | ((v2 == v0) & (i2 < i0));
        s_val[t] = take ? v2 : v0;
        s_idx[t] = take ? i2 : i0;
      }
      __syncthreads();
    }
    if (t == 0) s_far = s_idx[0];
    __syncthreads();
  }
}

// ---------------------------------------------------------------------------
// Kernel 2: nearest-centroid assignment via V_WMMA_F32_16X16X4_F32.
//   A (16x4, M=centroids): rows (-2cx,-2cy,-2cz,|c|^2)
//   B (4x16, N=points)   : cols (px,py,pz,1)
//   D[m][n] = -2 c.p + |c|^2  (squared distance minus per-point constant)
// One wave handles 16 points x 64 centroids with 4 WMMAs; per-point argmin is
// a branchless in-lane scan over 8 accumulator VGPRs x 4 tiles + one
// shfl_xor(16) merge. In-lane candidates arrive in increasing centroid index,
// so the lowest-index tie rule needs an index compare only at the merge.
// ---------------------------------------------------------------------------
__global__ __launch_bounds__(256)
void assign_wmma_kernel(const float* __restrict__ xyz,
                        const float* __restrict__ cdata,
                        float* __restrict__ out_clusters,
                        int N, int tiles_per_batch, long total_waves) {
  const long gwave = (long)(((size_t)blockIdx.x * blockDim.x + threadIdx.x) >> 5);
  if (gwave >= total_waves) return;  // uniform per wave: EXEC stays all-1s for WMMA

  const int lane = threadIdx.x & 31;
  const int half = lane >> 4;     // 0: K=0,1 rows, 1: K=2,3 rows
  const int l    = lane & 15;

  const int b    = (int)(gwave / tiles_per_batch);
  const int tile = (int)(gwave - (long)b * tiles_per_batch);

  int pi = tile * 16 + l;
  const bool valid = pi < N;
  if (pi >= N) pi = N - 1;
  const size_t po = 3 * ((size_t)b * N + (size_t)pi);
  const float px = xyz[po + 0], py = xyz[po + 1], pz = xyz[po + 2];

  // B-matrix lane fragment: lanes 0-15 hold (B[0][n],B[1][n]) = (px,py),
  // lanes 16-31 hold (B[2][n],B[3][n]) = (pz,1)
  v2f bmat;
  bmat.x = half ? pz : px;
  bmat.y = half ? 1.0f : py;

  const float4* __restrict__ c4 =
      reinterpret_cast<const float4*>(cdata) + (size_t)b * NPOINT;

  float best = 3.4e38f;
  int   bidx = 0x7fffffff;

#pragma unroll
  for (int tl = 0; tl < 4; ++tl) {
    float4 c = c4[tl * 16 + l];
    // A-matrix lane fragment: lanes 0-15 hold (A[m][0],A[m][1]),
    // lanes 16-31 hold (A[m][2],A[m][3])
    v2f amat;
    amat.x = half ? c.z : c.x;
    amat.y = half ? c.w : c.y;
    v8f acc = {};
    acc = __builtin_amdgcn_wmma_f32_16x16x4_f32(
        /*neg_a=*/false, amat, /*neg_b=*/false, bmat,
        /*c_mod=*/(short)0, acc, /*reuse_a=*/false, /*reuse_b=*/false);
#pragma unroll
    for (int r = 0; r < 8; ++r) {
      // D layout: VGPR r -> M=r (lanes 0-15), M=r+8 (lanes 16-31)
      int cidx = tl * 16 + half * 8 + r;
      float v = acc[r];
      // branchless: cidx strictly increases, so strict '<' keeps first-min
      bidx = (v < best) ? cidx : bidx;
      best = fminf(best, v);
    }
  }

  // merge the two lane-halves (other half covers the other 8 rows per tile)
  {
    float ov = __shfl_xor(best, 16, 32);
    int   oi = __shfl_xor(bidx, 16, 32);
    bool take = (ov < best) | ((ov == best) & (oi < bidx));
    bidx = take ? oi : bidx;
    best = take ? ov : best;
  }

  if (half == 0 && valid)
    out_clusters[(size_t)b * N + (size_t)tile * 16 + l] = (float)bidx;
}

// ---------------------------------------------------------------------------
extern "C" void kernel_launch(void* const* d_in, const int* in_sizes, int n_in,
                              void* d_out, int out_size, void* d_ws, size_t ws_size,
                              hipStream_t stream) {
  const float* xyz = (const float*)d_in[0];
  float* out = (float*)d_out;
  float* ws  = (float*)d_ws;

  // in_sizes[0] = B*N*3 ; out_size = B*(64*3 + 64 + N) = B*256 + B*N
  const int bn = in_sizes[0] / 3;
  int B = (out_size - bn) / 256;
  if (B <= 0) B = 1;
  const int N = bn / B;

  fps_kernel<<<B, FPS_THREADS, 0, stream>>>(xyz, out, ws, N, B);

  const int  tiles       = (N + 15) / 16;
  const long total_waves = (long)B * tiles;
  const int  blocks      = (int)((total_waves + 7) / 8);  // 8 waves per 256-thread block
  const float* cdata        = ws + (size_t)B * N;
  float*       out_clusters = out + (size_t)B * NPOINT * 3 + (size_t)B * NPOINT;

  assign_wmma_kernel<<<blocks, 256, 0, stream>>>(xyz, cdata, out_clusters,
                                                 N, tiles, total_waves);
}